// NMT_59313498357959
// MI455X (gfx1250) — compile-verified
//
#include <hip/hip_runtime.h>
#include <math.h>

// Problem constants from the reference
#define BB 64
#define SS 2048
#define HH 512

typedef __attribute__((ext_vector_type(2))) float v2f;
typedef __attribute__((ext_vector_type(8))) float v8f;

// ---------------------------------------------------------------------------
// GEMM: D[M x N] = A[M x K] * Bm[N x K]^T   (i.e. A @ Bm.T), optional tanh.
// One wave (32 threads) per 16x16 output tile, V_WMMA_F32_16X16X4_F32 K-loop.
// A-frag (16x4 f32): V0 = {K=0 lanes0-15, K=2 lanes16-31}, V1 = {K=1, K=3}
// B-frag (4x16 f32): V0 = {K=0 lanes0-15, K=2 lanes16-31}, V1 = {K=1, K=3},
//                    with N = lane%16;  B[k][n] = Bm[n][k] (contiguous pair).
// D-frag: VGPR r -> row M0+r (lanes0-15), row M0+r+8 (lanes16-31), N = lane%16.
// ---------------------------------------------------------------------------
__global__ __launch_bounds__(32) void nmt_gemm_nt_wmma(
    const float* __restrict__ A,   // [M x K]
    const float* __restrict__ Bm,  // [N x K]
    float* __restrict__ Dm,        // [M x N]
    int K, int N, int apply_tanh)
{
    const int mt   = blockIdx.x;
    const int nt   = blockIdx.y;
    const int lane = threadIdx.x & 31;
    const int hf   = lane >> 4;      // 0: lanes 0-15, 1: lanes 16-31
    const int l16  = lane & 15;
    const int M0 = mt * 16, N0 = nt * 16;

    const float* __restrict__ arow = A  + (size_t)(M0 + l16) * K + 2 * hf;
    const float* __restrict__ brow = Bm + (size_t)(N0 + l16) * K + 2 * hf;

    v8f c = {0.f, 0.f, 0.f, 0.f, 0.f, 0.f, 0.f, 0.f};
    for (int kk = 0; kk < K; kk += 4) {
        v2f a = *(const v2f*)(arow + kk);
        v2f b = *(const v2f*)(brow + kk);
        // (neg_a, A, neg_b, B, c_mod, C, reuse_a, reuse_b)
        c = __builtin_amdgcn_wmma_f32_16x16x4_f32(false, a, false, b,
                                                  (short)0, c, false, false);
    }
#pragma unroll
    for (int r = 0; r < 8; ++r) {
        float v = c[r];
        if (apply_tanh) v = tanhf(v);
        Dm[(size_t)(M0 + r + 8 * hf) * N + N0 + l16] = v;
    }
}

// ---------------------------------------------------------------------------
// pt = sigmoid(hidden_tanh[b,:] . w_pt); derive window center/left/right.
// One block per batch row.
// ---------------------------------------------------------------------------
__global__ __launch_bounds__(256) void nmt_pt_window_kernel(
    const float* __restrict__ hidden,   // [B x H], already tanh'ed
    const float* __restrict__ w_pt,     // [H]
    const int*   __restrict__ encode_len,
    const int*   __restrict__ Dptr,     // scalar window_size_d
    float* __restrict__ center_out,
    int*   __restrict__ left_out,
    int*   __restrict__ right_out)
{
    const int b = blockIdx.x;
    __shared__ float red[256];
    float acc = 0.f;
    for (int h = threadIdx.x; h < HH; h += 256)
        acc += hidden[(size_t)b * HH + h] * w_pt[h];
    red[threadIdx.x] = acc;
    __syncthreads();
    for (int s = 128; s > 0; s >>= 1) {
        if ((int)threadIdx.x < s) red[threadIdx.x] += red[threadIdx.x + s];
        __syncthreads();
    }
    if (threadIdx.x == 0) {
        const int Dw  = Dptr[0];
        const int len = encode_len[b];
        const float pt = 1.0f / (1.0f + expf(-red[0]));
        const float center = (float)len * pt;
        int c_int = (int)floorf(center);
        int left  = c_int - Dw; if (left < 0) left = 0;
        int right = c_int + Dw; if (right > len) right = len;
        center_out[b] = center;
        left_out[b]   = left;
        right_out[b]  = right;
    }
}

// ---------------------------------------------------------------------------
// Windowed attention: scores only for j in [left,right) (width <= 2*D),
// softmax over the window (exact equivalent of masked softmax over S),
// gaussian distance weighting, then ct[b,:] = sum_j at_j * encode_h[b,j,:].
// One block (8 waves) per batch row.
// ---------------------------------------------------------------------------
__global__ __launch_bounds__(256) void nmt_window_attn_kernel(
    const float* __restrict__ encode_h,  // [B x S x H]
    const float* __restrict__ yt,        // [B x H]
    const float* __restrict__ center_arr,
    const int*   __restrict__ left_arr,
    const int*   __restrict__ right_arr,
    const int*   __restrict__ Dptr,
    float* __restrict__ ct)              // [B x H]
{
    const int b = blockIdx.x;
    const int left  = left_arr[b];
    const int right = right_arr[b];
    const int width = right - left;      // <= 2*D (<= 2048 hard bound)
    const float center = center_arr[b];

    __shared__ float s_sc[2048];

    const int tid  = threadIdx.x;
    const int wave = tid >> 5;
    const int lane = tid & 31;

    const float* __restrict__ ytb = yt + (size_t)b * HH;
    const float* __restrict__ eb  = encode_h + (size_t)b * SS * HH;

    // Per-wave dot products: score_j = yt[b,:] . encode_h[b,j,:]
    for (int jj = wave; jj < width; jj += 8) {
        const float* __restrict__ row = eb + (size_t)(left + jj) * HH;
        float acc = 0.f;
        for (int h = lane; h < HH; h += 32) acc += ytb[h] * row[h];
        for (int off = 16; off > 0; off >>= 1) acc += __shfl_down(acc, off, 32);
        if (lane == 0) s_sc[jj] = acc;
    }
    __syncthreads();

    // Softmax over the window * gaussian weighting (serial, width <= 20)
    if (tid == 0) {
        const int Dw = Dptr[0];
        const float inv_den = 2.0f / ((float)Dw * (float)Dw);  // 1/(D^2/2)
        float m = -3.4e38f;
        for (int jj = 0; jj < width; ++jj) m = fmaxf(m, s_sc[jj]);
        float sum = 0.f;
        for (int jj = 0; jj < width; ++jj) {
            float e = expf(s_sc[jj] - m);
            s_sc[jj] = e;
            sum += e;
        }
        const float inv = 1.0f / sum;
        for (int jj = 0; jj < width; ++jj) {
            float d = (float)(left + jj) - center;
            s_sc[jj] = s_sc[jj] * inv * expf(-d * d * inv_den);
        }
    }
    __syncthreads();

    // ct[b,h] = sum_j at_j * encode_h[b, left+j, h]
    for (int h = tid; h < HH; h += 256) {
        float acc = 0.f;
        for (int jj = 0; jj < width; ++jj)
            acc += s_sc[jj] * eb[(size_t)(left + jj) * HH + h];
        ct[(size_t)b * HH + h] = acc;
    }
}

// ---------------------------------------------------------------------------
extern "C" void kernel_launch(void* const* d_in, const int* in_sizes, int n_in,
                              void* d_out, int out_size, void* d_ws, size_t ws_size,
                              hipStream_t stream) {
    const float* encode_h   = (const float*)d_in[0]; // [B,S,H]
    const int*   encode_len = (const int*)  d_in[1]; // [B]
    const float* yt         = (const float*)d_in[2]; // [B,H]
    const float* W_tan      = (const float*)d_in[3]; // [H,H]
    const float* w_pt       = (const float*)d_in[4]; // [H]
    const float* W_ct       = (const float*)d_in[5]; // [H,H]
    const int*   Dptr       = (const int*)  d_in[6]; // scalar D

    float* out = (float*)d_out;                      // [B,H]

    // Workspace layout
    char*  ws     = (char*)d_ws;
    float* hidden = (float*)ws;                                   // B*H floats
    float* ct     = (float*)(ws + (size_t)BB * HH * 4);           // B*H floats
    float* center = (float*)(ws + (size_t)2 * BB * HH * 4);       // B floats
    int*   left   = (int*)  (ws + (size_t)2 * BB * HH * 4 + BB * 4);
    int*   right  = (int*)  (ws + (size_t)2 * BB * HH * 4 + BB * 8);

    dim3 gemm_grid(BB / 16, HH / 16);  // (4, 32) 16x16 tiles, 1 wave each

    // 1) hidden = tanh(yt @ W_tan^T)
    nmt_gemm_nt_wmma<<<gemm_grid, 32, 0, stream>>>(yt, W_tan, hidden, HH, HH, 1);
    // 2) pt -> center / left / right
    nmt_pt_window_kernel<<<BB, 256, 0, stream>>>(hidden, w_pt, encode_len, Dptr,
                                                 center, left, right);
    // 3) windowed softmax * gaussian -> ct
    nmt_window_attn_kernel<<<BB, 256, 0, stream>>>(encode_h, yt, center, left,
                                                   right, Dptr, ct);
    // 4) ht = ct @ W_ct^T
    nmt_gemm_nt_wmma<<<gemm_grid, 32, 0, stream>>>(ct, W_ct, out, HH, HH, 0);
}